// FKPositionLoss_59631325938490
// MI455X (gfx1250) — compile-verified
//
#include <hip/hip_runtime.h>
#include <hip/hip_bf16.h>

#define NB 256
#define NJ 24
#define NT 1024
#define NF 8
#define NPART (NB * NF)   // 2048 partial sums

static __device__ constexpr int kParents[NJ] =
    {-1, 0, 0, 0, 1, 2, 3, 4, 5, 6, 7, 8, 9, 9, 9, 12, 13, 14, 16, 17, 18, 19, 20, 21};

struct Q4 { float w, x, y, z; };
struct V3 { float x, y, z; };

__device__ inline Q4 qmul(const Q4& a, const Q4& b) {
    Q4 r;
    r.w = a.w * b.w - a.x * b.x - a.y * b.y - a.z * b.z;
    r.x = a.w * b.x + a.x * b.w + a.y * b.z - a.z * b.y;
    r.y = a.w * b.y - a.x * b.z + a.y * b.w + a.z * b.x;
    r.z = a.w * b.z + a.x * b.y - a.y * b.x + a.z * b.w;
    return r;
}

__device__ inline V3 qrot(const Q4& q, V3 v) {
    // t = 2*cross(q.xyz, v); out = v + w*t + cross(q.xyz, t)
    float tx = 2.0f * (q.y * v.z - q.z * v.y);
    float ty = 2.0f * (q.z * v.x - q.x * v.z);
    float tz = 2.0f * (q.x * v.y - q.y * v.x);
    V3 r;
    r.x = v.x + q.w * tx + (q.y * tz - q.z * ty);
    r.y = v.y + q.w * ty + (q.z * tx - q.x * tz);
    r.z = v.z + q.w * tz + (q.x * ty - q.y * tx);
    return r;
}

__device__ inline Q4 sixd_to_quat(const float s[6]) {
    // Gram-Schmidt (rows b1,b2,b3) then branchless-equivalent matrix->quat.
    float n1 = sqrtf(s[0] * s[0] + s[1] * s[1] + s[2] * s[2]);
    n1 = fmaxf(n1, 1e-8f);
    float b1x = s[0] / n1, b1y = s[1] / n1, b1z = s[2] / n1;
    float d = b1x * s[3] + b1y * s[4] + b1z * s[5];
    float u2x = s[3] - d * b1x, u2y = s[4] - d * b1y, u2z = s[5] - d * b1z;
    float n2 = fmaxf(sqrtf(u2x * u2x + u2y * u2y + u2z * u2z), 1e-8f);
    float b2x = u2x / n2, b2y = u2y / n2, b2z = u2z / n2;
    float b3x = b1y * b2z - b1z * b2y;
    float b3y = b1z * b2x - b1x * b2z;
    float b3z = b1x * b2y - b1y * b2x;
    // R rows are b1,b2,b3
    float m00 = b1x, m01 = b1y, m02 = b1z;
    float m10 = b2x, m11 = b2y, m12 = b2z;
    float m20 = b3x, m21 = b3y, m22 = b3z;
    float q0 = sqrtf(fmaxf(1.0f + m00 + m11 + m22, 0.0f));
    float q1 = sqrtf(fmaxf(1.0f + m00 - m11 - m22, 0.0f));
    float q2 = sqrtf(fmaxf(1.0f - m00 + m11 - m22, 0.0f));
    float q3 = sqrtf(fmaxf(1.0f - m00 - m11 + m22, 0.0f));
    int best = 0; float bq = q0;
    if (q1 > bq) { best = 1; bq = q1; }
    if (q2 > bq) { best = 2; bq = q2; }
    if (q3 > bq) { best = 3; bq = q3; }
    float inv = 1.0f / (2.0f * fmaxf(bq, 0.1f));
    Q4 q;
    if (best == 0)      { q.w = q0 * q0;   q.x = m21 - m12; q.y = m02 - m20; q.z = m10 - m01; }
    else if (best == 1) { q.w = m21 - m12; q.x = q1 * q1;   q.y = m10 + m01; q.z = m02 + m20; }
    else if (best == 2) { q.w = m02 - m20; q.x = m10 + m01; q.y = q2 * q2;   q.z = m12 + m21; }
    else                { q.w = m10 - m01; q.x = m20 + m02; q.y = m21 + m12; q.z = q3 * q3;   }
    q.w *= inv; q.x *= inv; q.y *= inv; q.z *= inv;
    return q;
}

// Stage 1: one thread per (batch, frame). Full FK chain in registers
// (PARENTS is compile-time, parents precede children), write per-(b,f)
// sum of squared position error to d_ws.
__global__ void __launch_bounds__(256)
fk_loss_stage1(const float* __restrict__ pred, const float* __restrict__ targ,
               const float* __restrict__ root, const float* __restrict__ offs,
               float* __restrict__ partial) {
    int tid = blockIdx.x * blockDim.x + threadIdx.x;
    if (tid >= NPART) return;
    int f = tid & (NF - 1);
    int b = tid >> 3;
    int t = (f * (NT - 1)) / (NF - 1);  // matches int32(linspace(0,1023,8))

    V3 rt = { root[((size_t)b * 3 + 0) * NT + t],
              root[((size_t)b * 3 + 1) * NT + t],
              root[((size_t)b * 3 + 2) * NT + t] };
    const float* pb = pred + (size_t)b * NJ * 6 * NT + t;
    const float* tb = targ + (size_t)b * NJ * 6 * NT + t;

    V3 posp[NJ];
    {
        Q4 gq[NJ];
#pragma unroll
        for (int j = 0; j < NJ; ++j) {
            float s[6];
#pragma unroll
            for (int c = 0; c < 6; ++c) s[c] = pb[(size_t)(j * 6 + c) * NT];
            Q4 lq = sixd_to_quat(s);
            constexpr const int* P = kParents;
            int p = P[j];
            if (p < 0) { gq[j] = lq; posp[j] = rt; }
            else {
                V3 o = { offs[j * 3 + 0], offs[j * 3 + 1], offs[j * 3 + 2] };
                V3 dd = qrot(gq[p], o);
                posp[j] = { posp[p].x + dd.x, posp[p].y + dd.y, posp[p].z + dd.z };
                gq[j] = qmul(gq[p], lq);
            }
        }
    }

    float err = 0.0f;
    {
        Q4 gq[NJ];
        V3 post[NJ];
#pragma unroll
        for (int j = 0; j < NJ; ++j) {
            float s[6];
#pragma unroll
            for (int c = 0; c < 6; ++c) s[c] = tb[(size_t)(j * 6 + c) * NT];
            Q4 lq = sixd_to_quat(s);
            constexpr const int* P = kParents;
            int p = P[j];
            if (p < 0) { gq[j] = lq; post[j] = rt; }
            else {
                V3 o = { offs[j * 3 + 0], offs[j * 3 + 1], offs[j * 3 + 2] };
                V3 dd = qrot(gq[p], o);
                post[j] = { post[p].x + dd.x, post[p].y + dd.y, post[p].z + dd.z };
                gq[j] = qmul(gq[p], lq);
            }
            float dx = posp[j].x - post[j].x;
            float dy = posp[j].y - post[j].y;
            float dz = posp[j].z - post[j].z;
            err += dx * dx + dy * dy + dz * dz;
        }
    }
    partial[tid] = err;
}

// Stage 2: deterministic reduction of 2048 partials with V_WMMA_F32_16X16X4_F32.
// One wave32, EXEC all-ones, uniform control flow. B = all-ones so each WMMA
// adds 64 partials into the f32 accumulator tile; the total is invariant to
// the exact A lane layout (any bijection of 64 loads into A slots sums the same).
typedef float v2f_t __attribute__((ext_vector_type(2)));
typedef float v8f_t __attribute__((ext_vector_type(8)));

__global__ void __launch_bounds__(32)
fk_loss_reduce(const float* __restrict__ partial, float* __restrict__ out) {
    int lane = threadIdx.x;
    v8f_t acc = {};
    v2f_t ones;
    ones[0] = 1.0f; ones[1] = 1.0f;
    for (int i = 0; i < NPART; i += 64) {
        v2f_t a;
        a[0] = partial[i + lane * 2 + 0];
        a[1] = partial[i + lane * 2 + 1];
        acc = __builtin_amdgcn_wmma_f32_16x16x4_f32(
            /*neg_a=*/false, a, /*neg_b=*/false, ones,
            /*c_mod=*/(short)0, acc, /*reuse_a=*/false, /*reuse_b=*/false);
    }
    // C layout: lane<16 holds M=vgpr (col N=lane); lane>=16 holds M=vgpr+8 (col N=lane-16).
    // Every column holds identical values, so column 0 total = s(lane0) + s(lane16).
    float s = acc[0] + acc[1] + acc[2] + acc[3] + acc[4] + acc[5] + acc[6] + acc[7];
    float tot = __shfl(s, 0, 32) + __shfl(s, 16, 32);
    if (lane == 0) out[0] = tot * (1.0f / (float)(NB * NJ * 3 * NF));
}

extern "C" void kernel_launch(void* const* d_in, const int* in_sizes, int n_in,
                              void* d_out, int out_size, void* d_ws, size_t ws_size,
                              hipStream_t stream) {
    const float* pred = (const float*)d_in[0];  // (256,24,6,1024) f32
    const float* targ = (const float*)d_in[1];  // (256,24,6,1024) f32
    const float* root = (const float*)d_in[2];  // (256,3,1024)    f32
    const float* offs = (const float*)d_in[3];  // (24,3)          f32
    float* partial = (float*)d_ws;              // 2048 floats = 8 KB
    float* out = (float*)d_out;                 // scalar f32

    fk_loss_stage1<<<NPART / 256, 256, 0, stream>>>(pred, targ, root, offs, partial);
    fk_loss_reduce<<<1, 32, 0, stream>>>(partial, out);
}